// Filter__55113020342600
// MI455X (gfx1250) — compile-verified
//
#include <hip/hip_runtime.h>
#include <hip/hip_bf16.h>

// ---------------------------------------------------------------------------
// MMD-style kernel distance on MI455X (gfx1250, wave32).
// De: [8192,128] f32, X: [1024,128] f32, out: [1024] f32.
// Strategy: tiled fp32 GEMM on V_WMMA_F32_16X16X4_F32, fused multi-bandwidth
// Gaussian epilogue (v_exp_f32), symmetric-XX tile pruning (upper triangle
// only, off-diagonal tiles weighted 2x), deterministic hierarchical reduce.
// ---------------------------------------------------------------------------

typedef __attribute__((ext_vector_type(2))) float v2f;
typedef __attribute__((ext_vector_type(8))) float v8f;

#define NROWS  8192
#define BCOLS  1024
#define DDIM   128
#define BM     128
#define BN     128
#define BK     32
#define LDSS   33   // LDS row stride (floats); gcd(33,64)=1 -> conflict-free
#define NTILES (NROWS / BM)              // 64
#define NPAIRS (NTILES * (NTILES + 1) / 2) // 2080 upper-triangle tile pairs

// Gaussian coefficients: -0.5 / bw for bw in (10, 512, 2048)
#define C0 (-0.05f)
#define C1 (-9.765625e-4f)   // -0.5/512
#define C2 (-2.44140625e-4f) // -0.5/2048

__device__ __forceinline__ float kexp3(float d) {
    return __expf(C0 * d) + __expf(C1 * d) + __expf(C2 * d);
}

// ---------------------------------------------------------------------------
// Phase 1: squared row norms. One wave (32 lanes) per row; lane loads float4.
// ---------------------------------------------------------------------------
__global__ void norms_kernel(const float* __restrict__ De,
                             const float* __restrict__ X,
                             float* __restrict__ diag,
                             float* __restrict__ xsq) {
    int wave = (blockIdx.x * blockDim.x + threadIdx.x) >> 5;
    int lane = threadIdx.x & 31;
    if (wave >= NROWS + BCOLS) return;
    const float* row = (wave < NROWS) ? (De + (size_t)wave * DDIM)
                                      : (X + (size_t)(wave - NROWS) * DDIM);
    float4 v = *((const float4*)row + lane);
    float s = v.x * v.x + v.y * v.y + v.z * v.z + v.w * v.w;
    #pragma unroll
    for (int m = 16; m >= 1; m >>= 1) s += __shfl_xor(s, m, 32);
    if (lane == 0) {
        if (wave < NROWS) diag[wave] = s;
        else              xsq[wave - NROWS] = s;
    }
}

// ---------------------------------------------------------------------------
// Phase 2/3: tiled GEMM (A @ B^T over shared K=128) + fused exp epilogue.
// MODE 0 (XX): blockIdx.x is a linear upper-triangle tile-pair index; one
//              scalar partial per block (x2 weight for off-diagonal tiles).
// MODE 1 (XY): 2D grid; per-column partial sums over the tile's 128 A-rows.
// Block = 256 threads = 8 waves; wave grid 4 (M) x 2 (N); each wave computes
// a 32x64 patch = 2x4 subtiles of 16x16 driven by V_WMMA_F32_16X16X4_F32.
// ---------------------------------------------------------------------------
template <int MODE>
__global__ void gemm_mmd_kernel(const float* __restrict__ A,     // [*,128] rows
                                const float* __restrict__ Brows, // [*,128] rows
                                const float* __restrict__ dgAg,  // ||A_i||^2
                                const float* __restrict__ dgBg,  // ||B_j||^2
                                float* __restrict__ outP) {
    __shared__ float As[BM * LDSS];
    __shared__ float Bs[BN * LDSS];
    __shared__ float dgA[BM];
    __shared__ float dgB[BN];
    __shared__ float wsum[8];
    __shared__ float colacc[BN];

    const int tid  = threadIdx.x;
    const int lane = tid & 31;
    const int w    = tid >> 5;
    const int m16  = lane & 15;
    const int hi   = lane >> 4;           // 0: lanes 0-15, 1: lanes 16-31

    // Tile coordinates.
    int ti, tj;
    if (MODE == 0) {
        // Decode linear upper-triangle pair index (row-major: ti<=tj).
        int rem = blockIdx.x;
        int r = 0;
        while (rem >= (NTILES - r)) { rem -= (NTILES - r); ++r; }
        ti = r;
        tj = r + rem;
    } else {
        ti = blockIdx.x;
        tj = blockIdx.y;
    }
    const int i0 = ti * BM;               // A-row tile base
    const int j0 = tj * BN;               // B-row tile base

    const int waveM = w & 3;              // 0..3 -> 32-row band
    const int waveN = w >> 2;             // 0..1 -> 64-col band
    const int smB0  = waveM * 32;

    if (tid < BM) dgA[tid] = dgAg[i0 + tid];
    if (tid < BN) dgB[tid] = dgBg[j0 + tid];
    if (MODE == 1 && tid < BN) colacc[tid] = 0.0f;

    v8f acc[2][4];
    #pragma unroll
    for (int s = 0; s < 2; ++s)
        #pragma unroll
        for (int t = 0; t < 4; ++t)
            acc[s][t] = (v8f){0.f, 0.f, 0.f, 0.f, 0.f, 0.f, 0.f, 0.f};

    for (int kb = 0; kb < DDIM / BK; ++kb) {
        __syncthreads();
        // Stage 128x32 fp32 chunks of A and B into LDS (float4 global loads).
        #pragma unroll
        for (int e = tid; e < BM * (BK / 4); e += 256) {
            int m  = e >> 3;
            int k4 = e & 7;
            float4 va = *(const float4*)(A + (size_t)(i0 + m) * DDIM + kb * BK + k4 * 4);
            float* da = &As[m * LDSS + k4 * 4];
            da[0] = va.x; da[1] = va.y; da[2] = va.z; da[3] = va.w;
            float4 vb = *(const float4*)(Brows + (size_t)(j0 + m) * DDIM + kb * BK + k4 * 4);
            float* db = &Bs[m * LDSS + k4 * 4];
            db[0] = vb.x; db[1] = vb.y; db[2] = vb.z; db[3] = vb.w;
        }
        __syncthreads();

        #pragma unroll
        for (int k0 = 0; k0 < BK; k0 += 4) {
            // A fragments (16x4 fp32): lane = M, VGPR v = K {k0+v+2*hi}
            v2f a0, a1;
            {
                const float* p = &As[(smB0 + m16) * LDSS + k0 + 2 * hi];
                a0.x = p[0]; a0.y = p[1];
                const float* q = &As[(smB0 + 16 + m16) * LDSS + k0 + 2 * hi];
                a1.x = q[0]; a1.y = q[1];
            }
            // B fragments (4x16 fp32): lane = N, VGPR v = K {k0+v+2*hi}
            v2f bf[4];
            #pragma unroll
            for (int sn = 0; sn < 4; ++sn) {
                const float* p = &Bs[(waveN * 64 + sn * 16 + m16) * LDSS + k0 + 2 * hi];
                bf[sn].x = p[0]; bf[sn].y = p[1];
            }
            #pragma unroll
            for (int sn = 0; sn < 4; ++sn) {
                acc[0][sn] = __builtin_amdgcn_wmma_f32_16x16x4_f32(
                    false, a0, false, bf[sn], (short)0, acc[0][sn], false, false);
                acc[1][sn] = __builtin_amdgcn_wmma_f32_16x16x4_f32(
                    false, a1, false, bf[sn], (short)0, acc[1][sn], false, false);
            }
        }
    }
    __syncthreads();

    // Epilogue: d = ||a||^2 + ||b||^2 - 2*(a.b); k = sum_bw exp(-0.5/bw * d)
    float lsum = 0.0f;
    float csum[4] = {0.f, 0.f, 0.f, 0.f};
    #pragma unroll
    for (int sm = 0; sm < 2; ++sm) {
        const int mb = smB0 + sm * 16;
        #pragma unroll
        for (int sn = 0; sn < 4; ++sn) {
            const float dJ = dgB[waveN * 64 + sn * 16 + m16];
            #pragma unroll
            for (int v = 0; v < 8; ++v) {
                // C/D layout: VGPR v -> M = v (lanes 0-15) / v+8 (lanes 16-31)
                float dI = dgA[mb + v + 8 * hi];
                float d  = dI + dJ - 2.0f * acc[sm][sn][v];
                float kv = kexp3(d);
                if (MODE == 0) lsum += kv;
                else           csum[sn] += kv;
            }
        }
    }

    if (MODE == 0) {
        #pragma unroll
        for (int m = 16; m >= 1; m >>= 1) lsum += __shfl_xor(lsum, m, 32);
        if (lane == 0) wsum[w] = lsum;
        __syncthreads();
        if (tid == 0) {
            float t = 0.f;
            #pragma unroll
            for (int q = 0; q < 8; ++q) t += wsum[q];
            // Off-diagonal tiles stand in for their mirror image (kxx symm.)
            outP[blockIdx.x] = (ti == tj) ? t : 2.0f * t;
        }
    } else {
        // Lanes L and L+16 hold the same column N (different M halves).
        #pragma unroll
        for (int sn = 0; sn < 4; ++sn) {
            float c = csum[sn] + __shfl_xor(csum[sn], 16, 32);
            if (lane < 16) atomicAdd(&colacc[waveN * 64 + sn * 16 + lane], c);
        }
        __syncthreads();
        if (tid < BN)
            outP[(size_t)ti * (gridDim.y * BN) + tj * BN + tid] = colacc[tid];
    }
}

// ---------------------------------------------------------------------------
// Phase 4: deterministic reduction of the 2080 XX tile-pair partials.
// ---------------------------------------------------------------------------
__global__ void reduce_xx_kernel(const float* __restrict__ pxx,
                                 float* __restrict__ sxx) {
    __shared__ float sh[256];
    float s = 0.f;
    for (int i = threadIdx.x; i < NPAIRS; i += 256) s += pxx[i];
    sh[threadIdx.x] = s;
    __syncthreads();
    for (int st = 128; st >= 1; st >>= 1) {
        if (threadIdx.x < st) sh[threadIdx.x] += sh[threadIdx.x + st];
        __syncthreads();
    }
    if (threadIdx.x == 0) *sxx = sh[0];
}

// ---------------------------------------------------------------------------
// Phase 5: out[b] = mean_xx + 3 - 2*mean_xy[b]
// ---------------------------------------------------------------------------
__global__ void finalize_kernel(const float* __restrict__ colpart, // [64][1024]
                                const float* __restrict__ sxx,
                                float* __restrict__ out) {
    int b = blockIdx.x * blockDim.x + threadIdx.x;
    if (b >= BCOLS) return;
    float s = 0.f;
    #pragma unroll 8
    for (int t = 0; t < NROWS / BM; ++t) s += colpart[(size_t)t * BCOLS + b];
    float mean_xx = (*sxx) * (1.0f / ((float)NROWS * (float)NROWS));
    out[b] = mean_xx + 3.0f - 2.0f * (s * (1.0f / (float)NROWS));
}

// ---------------------------------------------------------------------------
// Workspace layout (floats):
//   [0 .. 8191]        diag    (||De_i||^2)
//   [8192 .. 9215]     xsq     (||X_b||^2)
//   [9216 .. 11295]    pxx     (2080 upper-triangle XX tile partials)
//   [11296 .. 76831]   colpart (64 row-tiles x 1024 columns)
//   [76832]            sxx     (scalar)
// Every slot is fully written each call -> no memset, deterministic.
// ---------------------------------------------------------------------------
extern "C" void kernel_launch(void* const* d_in, const int* in_sizes, int n_in,
                              void* d_out, int out_size, void* d_ws, size_t ws_size,
                              hipStream_t stream) {
    const float* De = (const float*)d_in[0]; // [8192,128]
    const float* X  = (const float*)d_in[1]; // [1024,128]
    float* out = (float*)d_out;              // [1024]

    float* ws      = (float*)d_ws;
    float* diag    = ws;
    float* xsq     = ws + 8192;
    float* pxx     = ws + 9216;
    float* colpart = ws + 11296;
    float* sxx     = ws + 76832;

    // Phase 1: row norms (9216 waves, 8 waves per 256-thread block)
    norms_kernel<<<(NROWS + BCOLS) / 8, 256, 0, stream>>>(De, X, diag, xsq);

    // Phase 2: XX term — upper-triangle tile pairs of the symmetric 8192^2
    // kernel matrix (2080 blocks instead of 4096: ~1.95x less compute).
    gemm_mmd_kernel<0><<<NPAIRS, 256, 0, stream>>>(De, De, diag, diag, pxx);

    // Phase 3: XY term — 64x8 tiles of the 8192x1024 cross matrix
    gemm_mmd_kernel<1><<<dim3(NROWS / BM, BCOLS / BN), 256, 0, stream>>>(
        De, X, diag, xsq, colpart);

    // Phase 4: reduce XX partials
    reduce_xx_kernel<<<1, 256, 0, stream>>>(pxx, sxx);

    // Phase 5: final per-sample distances
    finalize_kernel<<<BCOLS / 256, 256, 0, stream>>>(colpart, sxx, out);
}